// SpatialAttention_1382979470020
// MI455X (gfx1250) — compile-verified
//
#include <hip/hip_runtime.h>
#include <math.h>

// Problem sizes (fixed by the reference)
#define N_TOK  4096
#define DMODEL 256
#define NHEAD  4
#define HDIM   64
#define KNN    9      // k + 1 (include_self)

typedef __attribute__((ext_vector_type(2))) float v2f;
typedef __attribute__((ext_vector_type(4))) float v4f;
typedef __attribute__((ext_vector_type(8))) float v8f;

// ---------------------------------------------------------------------------
// LayerNorm: one 256-thread block per row, biased variance, eps=1e-5
// ---------------------------------------------------------------------------
__global__ __launch_bounds__(DMODEL) void ln_kernel(
    const float* __restrict__ x, const float* __restrict__ g,
    const float* __restrict__ b, float* __restrict__ out) {
  int row = blockIdx.x;
  int t = threadIdx.x;
  float v = x[row * DMODEL + t];
  __shared__ float red[DMODEL];
  red[t] = v;
  __syncthreads();
  for (int off = DMODEL / 2; off >= 1; off >>= 1) {
    if (t < off) red[t] += red[t + off];
    __syncthreads();
  }
  float mean = red[0] * (1.0f / DMODEL);
  __syncthreads();
  float d = v - mean;
  red[t] = d * d;
  __syncthreads();
  for (int off = DMODEL / 2; off >= 1; off >>= 1) {
    if (t < off) red[t] += red[t + off];
    __syncthreads();
  }
  float var = red[0] * (1.0f / DMODEL);
  float rs = rsqrtf(var + 1e-5f);
  out[row * DMODEL + t] = d * rs * g[t] + b[t];
}

// ---------------------------------------------------------------------------
// kNN (9 nearest incl. self): one 128-thread block per row.
// ---------------------------------------------------------------------------
#define KNN_T 128
__global__ __launch_bounds__(KNN_T) void knn_kernel(
    const float* __restrict__ coords, int* __restrict__ nbr) {
  int i = blockIdx.x;
  int t = threadIdx.x;
  float xi = coords[2 * i], yi = coords[2 * i + 1];
  float bd[KNN];
  int   bi[KNN];
#pragma unroll
  for (int s = 0; s < KNN; s++) { bd[s] = 3.4e38f; bi[s] = 0; }
  for (int j = t; j < N_TOK; j += KNN_T) {
    float dx = coords[2 * j] - xi;
    float dy = coords[2 * j + 1] - yi;
    float d2 = dx * dx + dy * dy;
    if (d2 < bd[KNN - 1]) {
      int p = KNN - 1;
      while (p > 0 && bd[p - 1] > d2) { bd[p] = bd[p - 1]; bi[p] = bi[p - 1]; p--; }
      bd[p] = d2; bi[p] = j;
    }
  }
  __shared__ float ld[KNN_T * KNN];
  __shared__ int   li[KNN_T * KNN];
  __shared__ int   ptr[KNN_T];
  __shared__ float rmin[KNN_T];
  __shared__ int   rarg[KNN_T];
#pragma unroll
  for (int s = 0; s < KNN; s++) { ld[t * KNN + s] = bd[s]; li[t * KNN + s] = bi[s]; }
  ptr[t] = 0;
  __syncthreads();
  for (int s = 0; s < KNN; s++) {
    int p = ptr[t];
    rmin[t] = (p < KNN) ? ld[t * KNN + p] : 3.4e38f;
    rarg[t] = t;
    __syncthreads();
    for (int off = KNN_T / 2; off >= 1; off >>= 1) {
      if (t < off && rmin[t + off] < rmin[t]) {
        rmin[t] = rmin[t + off];
        rarg[t] = rarg[t + off];
      }
      __syncthreads();
    }
    if (t == 0) {
      int w = rarg[0];
      nbr[i * KNN + s] = li[w * KNN + ptr[w]];
      ptr[w]++;
    }
    __syncthreads();
  }
}

// ---------------------------------------------------------------------------
// Tiled fp32 WMMA GEMM with async global->LDS double buffering.
//   C[M,Nd] = act(A[M,Kd] @ W[Nd,Kd]^T + bias) (+ resid)
// Block tile 64x64, 8 waves. K chunks of 32 streamed into LDS with
// GLOBAL_LOAD_ASYNC_TO_LDS_B128 (ASYNCcnt), double buffered so the DMA of
// chunk c+1 overlaps the V_WMMA_F32_16X16X4_F32 work on chunk c.
// LDS row stride 36 floats -> conflict-free 16-lane fragment reads.
// ---------------------------------------------------------------------------
#define BM 64
#define BN 64
#define BK 32
#define LDT 36

// LDS-aperture flat address carries the wave-relative LDS byte offset in [31:0]
__device__ __forceinline__ unsigned lds_off(const void* p) {
  return (unsigned)(unsigned long long)p;
}

#define ASYNC_B128(ldsoff, gaddr)                                   \
  asm volatile("global_load_async_to_lds_b128 %0, %1, off"          \
               :: "v"(ldsoff), "v"(gaddr) : "memory")

#define WAIT_ASYNC(n) \
  asm volatile("s_wait_asynccnt %0" :: "n"(n) : "memory")

__global__ __launch_bounds__(256) void gemm_kernel(
    const float* __restrict__ A, const float* __restrict__ W,
    const float* __restrict__ bias, const float* __restrict__ resid,
    float* __restrict__ C, int Kd, int Nd, int act) {
  __shared__ float As[2][BM * LDT];
  __shared__ float Bs[2][BN * LDT];
  int t = threadIdx.x;
  int wave = t >> 5, lane = t & 31;
  int mrow = wave & 3;      // 16-row band within block tile
  int ncol = wave >> 2;     // 32-col band within block tile
  int bm = blockIdx.y * BM;
  int bn = blockIdx.x * BN;
  int lm = lane & 15, lh = lane >> 4;

  v8f acc0 = {};
  v8f acc1 = {};

  // ---- async copy addressing: wave w owns rows [w*8, w*8+8) of both tiles,
  //      each instruction moves 4 rows (32 lanes x 16B).
  int crow = wave * 8 + (lane >> 3);   // tile row for instr 0 (instr 1: +4)
  int ccol = (lane & 7) * 4;           // float column within 32-wide chunk
  unsigned long long gA0 = (unsigned long long)(A + (size_t)(bm + crow) * Kd + ccol);
  unsigned long long gA1 = gA0 + (unsigned long long)4 * Kd * sizeof(float);
  unsigned long long gW0 = (unsigned long long)(W + (size_t)(bn + crow) * Kd + ccol);
  unsigned long long gW1 = gW0 + (unsigned long long)4 * Kd * sizeof(float);
  unsigned la0[2], la1[2], lb0[2], lb1[2];
#pragma unroll
  for (int b = 0; b < 2; b++) {
    la0[b] = lds_off(&As[b][crow * LDT + ccol]);
    la1[b] = lds_off(&As[b][(crow + 4) * LDT + ccol]);
    lb0[b] = lds_off(&Bs[b][crow * LDT + ccol]);
    lb1[b] = lds_off(&Bs[b][(crow + 4) * LDT + ccol]);
  }

  const int nch = Kd / BK;
  unsigned long long kbyte = 0;

  // prologue: stream chunk 0 into buffer 0
  ASYNC_B128(la0[0], gA0);
  ASYNC_B128(la1[0], gA1);
  ASYNC_B128(lb0[0], gW0);
  ASYNC_B128(lb1[0], gW1);

  for (int c = 0; c < nch; c++) {
    int cur = c & 1;
    if (c + 1 < nch) {
      // stream next chunk into the other buffer (safe: all waves finished
      // reading it at the trailing barrier of iteration c-1)
      unsigned long long kb = kbyte + BK * sizeof(float);
      int nxt = 1 - cur;
      ASYNC_B128(la0[nxt], gA0 + kb);
      ASYNC_B128(la1[nxt], gA1 + kb);
      ASYNC_B128(lb0[nxt], gW0 + kb);
      ASYNC_B128(lb1[nxt], gW1 + kb);
      WAIT_ASYNC(4);   // current chunk's 4 loads complete (in-order)
    } else {
      WAIT_ASYNC(0);
    }
    __syncthreads();   // publish LDS writes to all waves

    const float* arow  = &As[cur][(mrow * 16 + lm) * LDT + lh * 2];
    const float* brow0 = &Bs[cur][(ncol * 32 + lm) * LDT + lh * 2];
    const float* brow1 = brow0 + 16 * LDT;
#pragma unroll
    for (int kk = 0; kk < BK; kk += 4) {
      v2f a  = *(const v2f*)(arow + kk);
      v2f b0 = *(const v2f*)(brow0 + kk);
      v2f b1 = *(const v2f*)(brow1 + kk);
      acc0 = __builtin_amdgcn_wmma_f32_16x16x4_f32(
          false, a, false, b0, (short)0, acc0, false, false);
      acc1 = __builtin_amdgcn_wmma_f32_16x16x4_f32(
          false, a, false, b1, (short)0, acc1, false, false);
    }
    __syncthreads();   // all waves done reading buf[cur] before it is re-filled
    kbyte += BK * sizeof(float);
  }

  // Epilogue: bias (+gelu) (+residual), 16x16 C layout per ISA
  int row0 = bm + mrow * 16 + lh * 8;
  int col0 = bn + ncol * 32 + lm;
#pragma unroll
  for (int r = 0; r < 8; r++) {
    int row = row0 + r;
    {
      int col = col0;
      float v = acc0[r] + bias[col];
      if (act) v = 0.5f * v * (1.0f + erff(v * 0.70710678118f));
      if (resid) v += resid[(size_t)row * Nd + col];
      C[(size_t)row * Nd + col] = v;
    }
    {
      int col = col0 + 16;
      float v = acc1[r] + bias[col];
      if (act) v = 0.5f * v * (1.0f + erff(v * 0.70710678118f));
      if (resid) v += resid[(size_t)row * Nd + col];
      C[(size_t)row * Nd + col] = v;
    }
  }
}

// ---------------------------------------------------------------------------
// Sparse kNN attention: one block (4 heads x wave32) per row.
// ---------------------------------------------------------------------------
__global__ __launch_bounds__(NHEAD * 32) void attn_kernel(
    const float* __restrict__ Q, const float* __restrict__ Km,
    const float* __restrict__ V, const int* __restrict__ nbr,
    float* __restrict__ out) {
  int i = blockIdx.x;
  int t = threadIdx.x;
  int h = t >> 5, lane = t & 31;
  int base = h * HDIM + lane;
  float q0 = Q[(size_t)i * DMODEL + base];
  float q1 = Q[(size_t)i * DMODEL + base + 32];
  __shared__ int nb[KNN];
  if (t < KNN) nb[t] = nbr[i * KNN + t];
  __syncthreads();
  float s[KNN];
#pragma unroll
  for (int j = 0; j < KNN; j++) {
    int n = nb[j];
    float p = q0 * Km[(size_t)n * DMODEL + base] +
              q1 * Km[(size_t)n * DMODEL + base + 32];
#pragma unroll
    for (int off = 16; off >= 1; off >>= 1) p += __shfl_xor(p, off, 32);
    s[j] = p * 0.125f;  // 1/sqrt(64)
  }
  float m = s[0];
#pragma unroll
  for (int j = 1; j < KNN; j++) m = fmaxf(m, s[j]);
  float sum = 0.0f;
#pragma unroll
  for (int j = 0; j < KNN; j++) { s[j] = expf(s[j] - m); sum += s[j]; }
  float inv = 1.0f / sum;
  float o0 = 0.0f, o1 = 0.0f;
#pragma unroll
  for (int j = 0; j < KNN; j++) {
    int n = nb[j];
    float p = s[j] * inv;
    o0 += p * V[(size_t)n * DMODEL + base];
    o1 += p * V[(size_t)n * DMODEL + base + 32];
  }
  out[(size_t)i * DMODEL + base] = o0;
  out[(size_t)i * DMODEL + base + 32] = o1;
}

// ---------------------------------------------------------------------------
// Orchestration
// ---------------------------------------------------------------------------
extern "C" void kernel_launch(void* const* d_in, const int* in_sizes, int n_in,
                              void* d_out, int out_size, void* d_ws, size_t ws_size,
                              hipStream_t stream) {
  (void)in_sizes; (void)n_in; (void)out_size; (void)ws_size;
  const float* spot   = (const float*)d_in[0];
  const float* coords = (const float*)d_in[1];
  const float* Wq = (const float*)d_in[2];  const float* bq = (const float*)d_in[3];
  const float* Wk = (const float*)d_in[4];  const float* bk = (const float*)d_in[5];
  const float* Wv = (const float*)d_in[6];  const float* bv = (const float*)d_in[7];
  const float* Wo = (const float*)d_in[8];  const float* bo = (const float*)d_in[9];
  const float* g1 = (const float*)d_in[10]; const float* b1 = (const float*)d_in[11];
  const float* g2 = (const float*)d_in[12]; const float* b2 = (const float*)d_in[13];
  const float* Wf1 = (const float*)d_in[14]; const float* bf1 = (const float*)d_in[15];
  const float* Wf2 = (const float*)d_in[16]; const float* bf2 = (const float*)d_in[17];
  float* out = (float*)d_out;
  float* ws = (float*)d_ws;

  const size_t ND = (size_t)N_TOK * DMODEL;
  float* xn   = ws;            // [N,D]   (reused as h1 after x1 is formed)
  float* Qm   = ws + 1 * ND;   // [N,D]
  float* Km   = ws + 2 * ND;   // [N,D]
  float* Vm   = ws + 3 * ND;   // [N,D]
  float* attn = ws + 4 * ND;   // [N,D]
  float* x1   = ws + 5 * ND;   // [N,D]
  float* h1   = xn;            // reuse: xn dead after QKV+attn+x1
  float* hid  = Qm;            // reuse: [N,4D] spans Qm..attn (4*ND), dead by then
  int*   nbr  = (int*)(ws + 6 * ND);

  dim3 blk(256);
  dim3 gD(DMODEL / BN, N_TOK / BM);        // 256-wide outputs
  dim3 gF(4 * DMODEL / BN, N_TOK / BM);    // 1024-wide outputs

  ln_kernel<<<N_TOK, DMODEL, 0, stream>>>(spot, g1, b1, xn);
  knn_kernel<<<N_TOK, KNN_T, 0, stream>>>(coords, nbr);

  gemm_kernel<<<gD, blk, 0, stream>>>(xn, Wq, bq, nullptr, Qm, DMODEL, DMODEL, 0);
  gemm_kernel<<<gD, blk, 0, stream>>>(xn, Wk, bk, nullptr, Km, DMODEL, DMODEL, 0);
  gemm_kernel<<<gD, blk, 0, stream>>>(xn, Wv, bv, nullptr, Vm, DMODEL, DMODEL, 0);

  attn_kernel<<<N_TOK, NHEAD * 32, 0, stream>>>(Qm, Km, Vm, nbr, attn);

  gemm_kernel<<<gD, blk, 0, stream>>>(attn, Wo, bo, spot, x1, DMODEL, DMODEL, 0);
  ln_kernel<<<N_TOK, DMODEL, 0, stream>>>(x1, g2, b2, h1);
  gemm_kernel<<<gF, blk, 0, stream>>>(h1, Wf1, bf1, nullptr, hid, DMODEL, 4 * DMODEL, 1);
  gemm_kernel<<<gD, blk, 0, stream>>>(hid, Wf2, bf2, x1, out, 4 * DMODEL, DMODEL, 0);
}